// PointNet_19464791785585
// MI455X (gfx1250) — compile-verified
//
#include <hip/hip_runtime.h>

typedef __attribute__((ext_vector_type(2))) float v2f;
typedef __attribute__((ext_vector_type(4))) float f4;
typedef __attribute__((ext_vector_type(8))) float v8f;

#define SEQ 256
#define HID 1024
#define RNH 512

// ---------------------------------------------------------------------------
// Kernel 1: gates[s] = sigmoid( en[target[s]] . Wg[0:1024] + en[s] . Wg[1024:2048] )
// ---------------------------------------------------------------------------
__global__ void gates_kernel(const float* __restrict__ en,
                             const int*   __restrict__ target,
                             const float* __restrict__ Wg,
                             float*       __restrict__ gates) {
    __shared__ float red[256];
    const int s = blockIdx.x;
    const int t = threadIdx.x;
    const int tgt = target[s];
    const float* e_s = en + s   * HID;
    const float* e_t = en + tgt * HID;
    float acc = 0.f;
    for (int h = t; h < HID; h += 256)
        acc += e_t[h] * Wg[h] + e_s[h] * Wg[HID + h];
    red[t] = acc;
    __syncthreads();
    for (int off = 128; off > 0; off >>= 1) {
        if (t < off) red[t] += red[t + off];
        __syncthreads();
    }
    if (t == 0) gates[s] = 1.0f / (1.0f + __expf(-red[0]));
}

// ---------------------------------------------------------------------------
// Kernel 2: three 256xK @ KxN f32 GEMMs via V_WMMA_F32_16X16X4_F32.
//   z==0: EA  = en @ W1^T             (K=1024)
//   z==1: EBg = (en[target] @ W1^T) * gates[row]  (K=1024, gate folded in epilogue)
//   z==2: WD  = de @ W2^T             (K=512)
// One wave per 16x16 output tile; 4 waves (4 N-tiles) per block.
// A-operand lane layout (ISA 7.12.2, 32-bit A 16x4): lane l -> M=l&15,
//   K = koff,koff+1 with koff=(l>>4)*2.  B-operand symmetric with N=l&15,
//   which is exactly a row of W1/W2 (giving the transpose for free).
// C/D layout: VGPR v, lane l -> M = v + (l>>4)*8, N = l&15.
// ---------------------------------------------------------------------------
__global__ void gemm_wmma_kernel(const float* __restrict__ en,
                                 const float* __restrict__ de,
                                 const int*   __restrict__ target,
                                 const float* __restrict__ W1,
                                 const float* __restrict__ W2,
                                 const float* __restrict__ gates,
                                 float* __restrict__ EA,
                                 float* __restrict__ EBg,
                                 float* __restrict__ WD) {
    const int lane = threadIdx.x & 31;
    const int wave = threadIdx.x >> 5;
    const int z  = blockIdx.z;
    const int n0 = (blockIdx.x * 4 + wave) * 16;
    const int m0 = blockIdx.y * 16;

    const int mrow = lane & 15;          // M (for A) and N (for B) lane index
    const int koff = (lane >> 4) * 2;    // K sub-offset within the 16x16x4 step

    const float* Arow;
    const float* Brow;
    float* Out;
    int K;
    if (z == 0) {
        Arow = en + (size_t)(m0 + mrow) * HID;
        Brow = W1 + (size_t)(n0 + mrow) * HID;
        Out = EA;  K = HID;
    } else if (z == 1) {
        Arow = en + (size_t)target[m0 + mrow] * HID;
        Brow = W1 + (size_t)(n0 + mrow) * HID;
        Out = EBg; K = HID;
    } else {
        Arow = de + (size_t)(m0 + mrow) * RNH;
        Brow = W2 + (size_t)(n0 + mrow) * RNH;
        Out = WD;  K = RNH;
    }

    v8f c = {};
    for (int k = 0; k < K; k += 4) {
        v2f a = *(const v2f*)(Arow + k + koff);   // 8B aligned: k%4==0, koff in {0,2}
        v2f b = *(const v2f*)(Brow + k + koff);
        // 8 args: (neg_a, A, neg_b, B, c_mod, C, reuse_a, reuse_b)
        c = __builtin_amdgcn_wmma_f32_16x16x4_f32(false, a, false, b,
                                                  (short)0, c, false, false);
    }

    const int crow = (lane >> 4) * 8;
    const int ccol = lane & 15;
    if (z == 1) {
#pragma unroll
        for (int v = 0; v < 8; v++) c[v] *= gates[m0 + crow + v];
    }
#pragma unroll
    for (int v = 0; v < 8; v++)
        Out[(size_t)(m0 + crow + v) * RNH + n0 + ccol] = c[v];
}

// ---------------------------------------------------------------------------
// Kernel 3: att[i,j] = sum_r relu(EA[j,r] + mask[i,j]*EBg[j,r] + WD[i,r]) * V[r]
// 16x16 (i,j) tile per block, 256 threads, LDS-staged 128-wide r-chunks.
// Row stride 136 floats -> float4 LDS stores stay 16B-aligned (544 % 16 == 0).
// ---------------------------------------------------------------------------
__global__ void att_kernel(const float* __restrict__ mask,
                           const float* __restrict__ V,
                           const float* __restrict__ EA,
                           const float* __restrict__ EBg,
                           const float* __restrict__ WD,
                           float* __restrict__ out) {
    __shared__ float As[16][136];
    __shared__ float Bs[16][136];
    __shared__ float Ws[16][136];
    __shared__ float Vs[128];

    const int tid = threadIdx.x;
    const int ti = tid >> 4;   // local i
    const int tj = tid & 15;   // local j
    const int i0 = blockIdx.y * 16;
    const int j0 = blockIdx.x * 16;

    const float m = mask[(size_t)(i0 + ti) * SEQ + (j0 + tj)];
    float acc = 0.f;

    for (int r0 = 0; r0 < RNH; r0 += 128) {
#pragma unroll
        for (int p = 0; p < 2; p++) {
            const int idx = (p * 256 + tid) * 4;   // 2048 floats / matrix / chunk
            const int row = idx >> 7;
            const int col = idx & 127;
            f4 va = *(const f4*)(EA  + (size_t)(j0 + row) * RNH + r0 + col);
            f4 vb = *(const f4*)(EBg + (size_t)(j0 + row) * RNH + r0 + col);
            f4 vw = *(const f4*)(WD  + (size_t)(i0 + row) * RNH + r0 + col);
            *(f4*)&As[row][col] = va;
            *(f4*)&Bs[row][col] = vb;
            *(f4*)&Ws[row][col] = vw;
        }
        if (tid < 128) Vs[tid] = V[r0 + tid];
        __syncthreads();

#pragma unroll 4
        for (int r = 0; r < 128; r++) {
            const float t = As[tj][r] + m * Bs[tj][r] + Ws[ti][r];
            acc += fmaxf(t, 0.f) * Vs[r];
        }
        __syncthreads();
    }
    out[(size_t)(i0 + ti) * SEQ + (j0 + tj)] = acc;
}

// ---------------------------------------------------------------------------
extern "C" void kernel_launch(void* const* d_in, const int* in_sizes, int n_in,
                              void* d_out, int out_size, void* d_ws, size_t ws_size,
                              hipStream_t stream) {
    const float* en     = (const float*)d_in[0];   // (1,256,1024)
    const float* de     = (const float*)d_in[1];   // (1,256,512)
    const int*   target = (const int*)  d_in[2];   // (256,)
    const float* mask   = (const float*)d_in[3];   // (256,256)
    const float* W1     = (const float*)d_in[4];   // (512,1024)
    const float* W2     = (const float*)d_in[5];   // (512,512)
    const float* V      = (const float*)d_in[6];   // (512,)
    const float* Wg     = (const float*)d_in[7];   // (2048,)
    float* out = (float*)d_out;                    // (1,256,256)

    float* gates = (float*)d_ws;                   // 256
    float* EA    = gates + SEQ;                    // 256*512
    float* EBg   = EA + SEQ * RNH;                 // 256*512
    float* WD    = EBg + SEQ * RNH;                // 256*512  (~1.5 MB total)

    gates_kernel<<<SEQ, 256, 0, stream>>>(en, target, Wg, gates);
    gemm_wmma_kernel<<<dim3(RNH / 64, SEQ / 16, 3), 128, 0, stream>>>(
        en, de, target, W1, W2, gates, EA, EBg, WD);
    att_kernel<<<dim3(SEQ / 16, SEQ / 16), 256, 0, stream>>>(
        mask, V, EA, EBg, WD, out);
}